// GNNModel_49675591745836
// MI455X (gfx1250) — compile-verified
//
#include <hip/hip_runtime.h>
#include <cstddef>

typedef __attribute__((ext_vector_type(2))) float v2f;
typedef __attribute__((ext_vector_type(8))) float v8f;

// ---------------------------------------------------------------------------
// Degree / normalization: deg[i] = 1 (self loop) + #edges whose source (ei row 0) is i; dinv = deg^-1/2
// ---------------------------------------------------------------------------
__global__ void deg_init_k(float* __restrict__ deg, int n) {
  int i = blockIdx.x * blockDim.x + threadIdx.x;
  if (i < n) deg[i] = 1.0f;  // self-loop contribution
}

__global__ void deg_count_k(const int* __restrict__ col, float* __restrict__ deg, int e) {
  int i = blockIdx.x * blockDim.x + threadIdx.x;
  if (i < e) unsafeAtomicAdd(&deg[col[i]], 1.0f);  // native global_atomic_add_f32
}

__global__ void dinv_k(float* __restrict__ deg, int n) {
  int i = blockIdx.x * blockDim.x + threadIdx.x;
  if (i < n) deg[i] = rsqrtf(deg[i]);  // deg >= 1 always
}

// ---------------------------------------------------------------------------
// Dense GEMM: C[n_rows x NCOLS] = (RELU_A ? relu(A) : A)[n_rows x K] @ W[K x NCOLS] (+bias)
// via V_WMMA_F32_16X16X4_F32.
//
//  * W is staged once per block into LDS *transposed* with row padding:
//    Ws[col][k], KP = K+4. A B-fragment (k,k+1) at fixed col is then one
//    contiguous 8-byte-aligned pair -> single ds_load_b64 straight into an
//    even VGPR pair (no packing movs). Bank index = 4*col + k (mod 64):
//    lanes 0-15 (kb=0) and 16-31 (kb=2) together hit all 64 banks once.
//  * Each wave owns 16 rows x NCOLS: A fragment loaded once per k-step
//    (single 8-byte global load) and reused across NCOLS/16 WMMAs.
//
// Fragment layouts (ISA 7.12.2, 32-bit):
//   A 16x4 : lanes 0-15 -> M=lane, {v0,v1}={K0,K1}; lanes 16-31 -> M=lane-16, {K2,K3}
//   B 4x16 : lanes 0-15 -> N=lane, {v0,v1}={K0,K1}; lanes 16-31 -> N=lane-16, {K2,K3}
//   C/D    : vr, lanes 0-15 -> (M=r, N=lane); lanes 16-31 -> (M=8+r, N=lane-16)
// ---------------------------------------------------------------------------
template <bool RELU_A, int NCOLS, int K = 64>
__global__ void gemm_wmma_k(const float* __restrict__ A, const float* __restrict__ W,
                            const float* __restrict__ bias, float* __restrict__ C,
                            int n_rows) {
  constexpr int CT = NCOLS / 16;  // column tiles per wave
  constexpr int KP = K + 4;       // padded k-stride: conflict-free b64 reads
  __shared__ float Ws[NCOLS * KP];

  // Cooperative transposed stage of W into LDS.
  for (int i = threadIdx.x; i < K * NCOLS; i += blockDim.x) {
    int r = i / NCOLS;  // k   (NCOLS is a compile-time power of two)
    int c = i % NCOLS;  // col
    Ws[c * KP + r] = W[i];
  }
  __syncthreads();

  const int lane = threadIdx.x & 31;
  const int wv = threadIdx.x >> 5;
  const int row_tile = blockIdx.x * (blockDim.x >> 5) + wv;
  if (row_tile * 16 >= n_rows) return;  // wave-uniform: EXEC all-1s below
  const int row0 = row_tile << 4;

  const int mn = lane & 15;         // M for A-frag, N for B-frag / C-D columns
  const int kb = (lane >> 4) << 1;  // K sub-offset: 0 (lanes 0-15) or 2 (lanes 16-31)

  int arow = row0 + mn;
  if (arow >= n_rows) arow = n_rows - 1;  // clamp; tail rows masked at store
  const float* __restrict__ Ap = A + (size_t)arow * K + kb;
  const float* __restrict__ Bp = &Ws[mn * KP + kb];  // +t*16*KP +k0 are DS immediates

  v8f acc[CT];
#pragma unroll
  for (int t = 0; t < CT; ++t) acc[t] = (v8f){0.f, 0.f, 0.f, 0.f, 0.f, 0.f, 0.f, 0.f};

#pragma unroll
  for (int k0 = 0; k0 < K; k0 += 4) {
    v2f a = *(const v2f*)(Ap + k0);  // 8B aligned: global_load_b64
    if (RELU_A) { a.x = fmaxf(a.x, 0.0f); a.y = fmaxf(a.y, 0.0f); }
#pragma unroll
    for (int t = 0; t < CT; ++t) {
      v2f b = *(const v2f*)(Bp + t * 16 * KP + k0);  // single ds_load_b64
      // (neg_a, A, neg_b, B, c_mod, C, reuse_a, reuse_b)
      acc[t] = __builtin_amdgcn_wmma_f32_16x16x4_f32(false, a, false, b, (short)0,
                                                     acc[t], false, false);
    }
  }

  const int mb = (lane >> 4) << 3;  // 0 or 8
#pragma unroll
  for (int t = 0; t < CT; ++t) {
    const int cidx = t * 16 + mn;
    const float bv = bias ? bias[cidx] : 0.0f;
#pragma unroll
    for (int r = 0; r < 8; ++r) {
      int rr = row0 + mb + r;
      if (rr < n_rows) C[(size_t)rr * NCOLS + cidx] = acc[t][r] + bv;
    }
  }
}

// ---------------------------------------------------------------------------
// Aggregation init: h[i,c] = t[i,c] * dinv[i]^2 (self-loop message) + bias[c]
// Channel count fixed at 64 (shifts, no integer division).
// ---------------------------------------------------------------------------
__global__ void self_init_k(const float* __restrict__ t, const float* __restrict__ dinv,
                            const float* __restrict__ bias, float* __restrict__ h,
                            int n) {
  int idx = blockIdx.x * blockDim.x + threadIdx.x;
  if (idx >= (n << 6)) return;
  int i = idx >> 6;
  int ch = idx & 63;
  float d = dinv[i];
  h[idx] = t[idx] * (d * d) + bias[ch];
}

// ---------------------------------------------------------------------------
// Edge scatter: one wave per edge; 64 channels -> 2 floats per lane.
// h[col] += t[row] * (dinv[row]*dinv[col])  via fp32 global atomics.
// ---------------------------------------------------------------------------
__global__ void edge_scatter_k(const int* __restrict__ ei, const float* __restrict__ dinv,
                               const float* __restrict__ t, float* __restrict__ h, int e) {
  const int lane = threadIdx.x & 31;
  const int w = (blockIdx.x * blockDim.x + threadIdx.x) >> 5;
  if (w >= e) return;
  const int r = ei[w];       // source node
  const int c = ei[e + w];   // target node
  const float nrm = dinv[r] * dinv[c];
  const float* __restrict__ src = t + ((size_t)r << 6);
  float* __restrict__ dst = h + ((size_t)c << 6);
  unsafeAtomicAdd(&dst[lane],      src[lane]      * nrm);
  unsafeAtomicAdd(&dst[lane + 32], src[lane + 32] * nrm);
}

// ---------------------------------------------------------------------------
extern "C" void kernel_launch(void* const* d_in, const int* in_sizes, int n_in,
                              void* d_out, int out_size, void* d_ws, size_t ws_size,
                              hipStream_t stream) {
  const float* x    = (const float*)d_in[0];
  const int*   ei   = (const int*)d_in[1];
  const float* W1   = (const float*)d_in[2];
  const float* b1   = (const float*)d_in[3];
  const float* W2   = (const float*)d_in[4];
  const float* b2   = (const float*)d_in[5];
  const float* Wlin = (const float*)d_in[6];
  const float* blin = (const float*)d_in[7];
  float* out = (float*)d_out;

  const int IN_C = 64, HID_C = 64;
  const int N = in_sizes[0] / IN_C;
  const int E = in_sizes[1] / 2;

  // Workspace: dinv [N], T [N*64] (post-GEMM), H [N*64] (aggregated). ~52 MB.
  float* dinv = (float*)d_ws;
  size_t npad = ((size_t)N + 63) & ~(size_t)63;
  float* T = dinv + npad;
  float* H = T + (size_t)N * HID_C;

  const int TB = 256;  // 8 waves per block (wave32)
  auto cdiv = [](long long a, long long b) { return (int)((a + b - 1) / b); };

  // --- symmetric normalization ---
  deg_init_k<<<cdiv(N, TB), TB, 0, stream>>>(dinv, N);
  deg_count_k<<<cdiv(E, TB), TB, 0, stream>>>(ei, dinv, E);
  dinv_k<<<cdiv(N, TB), TB, 0, stream>>>(dinv, N);

  const int row_tiles = (N + 15) / 16;
  const int gemm_blocks = cdiv(row_tiles, TB / 32);  // one 16-row strip per wave

  // --- layer 1: T = x@W1 ; H = scatter(T)+b1 (relu deferred into next GEMM) ---
  gemm_wmma_k<false, 64><<<gemm_blocks, TB, 0, stream>>>(x, W1, nullptr, T, N);
  self_init_k<<<cdiv((long long)N * 64, TB), TB, 0, stream>>>(T, dinv, b1, H, N);
  edge_scatter_k<<<cdiv((long long)E * 32, TB), TB, 0, stream>>>(ei, dinv, T, H, E);

  // --- layer 2: T = relu(H)@W2 ; H = scatter(T)+b2 ---
  gemm_wmma_k<true, 64><<<gemm_blocks, TB, 0, stream>>>(H, W2, nullptr, T, N);
  self_init_k<<<cdiv((long long)N * 64, TB), TB, 0, stream>>>(T, dinv, b2, H, N);
  edge_scatter_k<<<cdiv((long long)E * 32, TB), TB, 0, stream>>>(ei, dinv, T, H, E);

  // --- head: out = relu(H)@Wlin + blin ---
  gemm_wmma_k<true, 32><<<gemm_blocks, TB, 0, stream>>>(H, Wlin, blin, out, N);
}